// net_57767310131841
// MI455X (gfx1250) — compile-verified
//
#include <hip/hip_runtime.h>
#include <hip/hip_bf16.h>

// ---------------------------------------------------------------------------
// Problem constants (match reference setup_inputs)
// ---------------------------------------------------------------------------
#define C_CH   256
#define POOL   7
#define PP     (POOL * POOL)     // 49
#define NLEV   4
#define NCLS   3
#define MH     28
#define MW     28
#define MHW    (MH * MW)         // 784

typedef float v2f __attribute__((ext_vector_type(2)));
typedef float v8f __attribute__((ext_vector_type(8)));

// Per-ROI precomputed sampling parameters (46 dwords)
struct RoiParams {
    int   level;
    int   bix;
    int   H;        // H == W for square pyramid levels
    int   pad;
    int   y0[POOL];
    int   y1[POOL];
    int   x0[POOL];
    int   x1[POOL];
    float wy[POOL];
    float wx[POOL];
};

// ---------------------------------------------------------------------------
// Wave-wide sum using V_WMMA_F32_16X16X4_F32 as a horizontal-reduction engine.
// f32 A layout (ISA 7.12.2): lanes 0-15 hold {K0,K1} for M=lane,
// lanes 16-31 hold {K2,K3} for M=lane-16.  With A = (x, 0) per lane and
// B = all-ones (layout independent), D[m][*] = x_m + x_{m+16}.
// D layout: VGPR r -> row r (lanes 0-15) / row r+8 (lanes 16-31), so summing
// the 8 D registers and one shfl_xor(16) yields the full 32-lane sum in every
// lane.  EXEC must be all ones: call unconditionally from converged code.
// ---------------------------------------------------------------------------
__device__ __forceinline__ float wave_sum_wmma(float x) {
    v2f a; a.x = x;    a.y = 0.0f;
    v2f b; b.x = 1.0f; b.y = 1.0f;
    v8f c = {};
    v8f d = __builtin_amdgcn_wmma_f32_16x16x4_f32(
        /*neg_a=*/false, a, /*neg_b=*/false, b,
        /*c_mod=*/(short)0, c, /*reuse_a=*/false, /*reuse_b=*/false);
    float s = d[0] + d[1] + d[2] + d[3] + d[4] + d[5] + d[6] + d[7];
    s += __shfl_xor(s, 16, 32);     // combine the two lane-halves (rows 0-7 vs 8-15)
    return s;
}

// ---------------------------------------------------------------------------
// Kernel 1: per-ROI level selection + bilinear sampling parameters
// ---------------------------------------------------------------------------
__global__ void roi_params_kernel(const float* __restrict__ rois,
                                  RoiParams* __restrict__ rp, int N) {
    int n = blockIdx.x * blockDim.x + threadIdx.x;
    if (n >= N) return;
    const float* r = rois + (size_t)n * 5;
    float y1 = r[0], x1 = r[1], y2 = r[2], x2 = r[3];
    int bix = (int)r[4];
    float h = y2 - y1, w = x2 - x1;
    float lvlf = rintf(4.0f + log2f(sqrtf(h * w) + 1e-12f));   // RNE == jnp.round
    int level = lvlf < 0.0f ? 0 : (lvlf > 3.0f ? 3 : (int)lvlf);
    int H = 128 >> level;
    float scale = (float)(H - 1);

    RoiParams p;
    p.level = level; p.bix = bix; p.H = H; p.pad = 0;
    #pragma unroll
    for (int i = 0; i < POOL; ++i) {
        float g  = (float)i / 6.0f;
        float ys = (y1 + g * (y2 - y1)) * scale;
        float xs = (x1 + g * (x2 - x1)) * scale;
        float yf = floorf(ys), xf = floorf(xs);
        p.wy[i] = ys - yf;
        p.wx[i] = xs - xf;
        float hm1 = scale;
        p.y0[i] = (int)fminf(fmaxf(yf,        0.0f), hm1);
        p.y1[i] = (int)fminf(fmaxf(yf + 1.0f, 0.0f), hm1);
        p.x0[i] = (int)fminf(fmaxf(xf,        0.0f), hm1);
        p.x1[i] = (int)fminf(fmaxf(xf + 1.0f, 0.0f), hm1);
    }
    rp[n] = p;
}

// ---------------------------------------------------------------------------
// Kernel 2: ROI-align gather.  One workgroup per ROI; params staged in LDS;
// 49 iterations x 256 threads cover all 256*49 output elements with fully
// coalesced stores.  Gathers hit L2 (whole pyramid fits in 192 MB).
// ---------------------------------------------------------------------------
__global__ void __launch_bounds__(256)
roi_align_kernel(const float* __restrict__ p2, const float* __restrict__ p3,
                 const float* __restrict__ p4, const float* __restrict__ p5,
                 const RoiParams* __restrict__ rp, float* __restrict__ out) {
    constexpr int RP_DW = sizeof(RoiParams) / 4;
    __shared__ RoiParams sp;
    const int n = blockIdx.x;
    const int t = threadIdx.x;

    {   // stage params into LDS
        const int* src = (const int*)&rp[n];
        int*       dst = (int*)&sp;
        if (t < RP_DW) dst[t] = src[t];
    }
    __syncthreads();

    const int H     = sp.H;
    const int plane = H * H;
    const float* fm = (sp.level == 0) ? p2 : (sp.level == 1) ? p3
                    : (sp.level == 2) ? p4 : p5;
    const float* base = fm + (size_t)sp.bix * C_CH * plane;
    float* outn = out + (size_t)n * C_CH * PP;

    #pragma unroll 1
    for (int i = 0; i < PP; ++i) {
        int e  = i * 256 + t;            // 0 .. 256*49-1
        int c  = e / PP;
        int p  = e - c * PP;
        int py = p / POOL;
        int px = p - py * POOL;

        const float* pl = base + (size_t)c * plane;
        int y0 = sp.y0[py], yb = sp.y1[py];
        int x0 = sp.x0[px], xb = sp.x1[px];
        float wy = sp.wy[py], wx = sp.wx[px];
        float v00 = pl[y0 * H + x0];
        float v01 = pl[y0 * H + xb];
        float v10 = pl[yb * H + x0];
        float v11 = pl[yb * H + xb];
        float omy = 1.0f - wy, omx = 1.0f - wx;
        outn[e] = v00 * omy * omx + v01 * omy * wx
                + v10 * wy  * omx + v11 * wy  * wx;
    }
}

// ---------------------------------------------------------------------------
// Kernel 3: per-ROI masked BCE partial sums (600 blocks x 256 threads),
// reduced per-wave through WMMA, cross-wave through LDS in fixed order.
// ---------------------------------------------------------------------------
__global__ void __launch_bounds__(256)
mask_partial_kernel(const int* __restrict__ tcid,
                    const float* __restrict__ pred_masks,
                    const float* __restrict__ target_masks,
                    float* __restrict__ partials) {
    const int r = blockIdx.x;
    const int t = threadIdx.x;
    const int tc = tcid[r];          // uniform per block

    float x = 0.0f;
    if (tc > 0) {                    // block-uniform branch; EXEC stays full
        const float* pm = pred_masks + ((size_t)r * NCLS + tc) * MHW;
        const float* tm = target_masks + (size_t)r * MHW;
        for (int j = t; j < MHW; j += 256) {
            float pv = fminf(fmaxf(pm[j], 1e-7f), 1.0f - 1e-7f);
            float tv = tm[j];
            x += -(tv * logf(pv) + (1.0f - tv) * logf(1.0f - pv));
        }
    }

    __shared__ float lw[8];
    float ws = wave_sum_wmma(x);
    if ((t & 31) == 0) lw[t >> 5] = ws;
    __syncthreads();
    if (t == 0) {
        float tot = 0.0f;
        #pragma unroll
        for (int i = 0; i < 8; ++i) tot += lw[i];
        partials[r] = tot;
    }
}

// ---------------------------------------------------------------------------
// Kernel 4: finalize the three losses in one 640-thread block (20 waves).
// ---------------------------------------------------------------------------
__global__ void __launch_bounds__(640)
losses_kernel(const int* __restrict__ tcid,
              const float* __restrict__ pred_deltas,   // (R,3,4)
              const float* __restrict__ tgt_deltas,    // (R,4)
              const float* __restrict__ logits,        // (R,3)
              const float* __restrict__ mask_partials, // (R)
              float* __restrict__ out3, int R) {
    const int t = threadIdx.x;
    float termC = 0.0f, termB = 0.0f, posf = 0.0f, termM = 0.0f;

    if (t < R) {
        int tc = tcid[t];
        // ---- class loss term: lse - logit[target]
        const float* lg = logits + (size_t)t * NCLS;
        float m   = fmaxf(lg[0], fmaxf(lg[1], lg[2]));
        float lse = m + logf(expf(lg[0] - m) + expf(lg[1] - m) + expf(lg[2] - m));
        termC = lse - lg[tc];
        // ---- bbox smooth-L1 (masked by pos => only compute for tc>0)
        if (tc > 0) {
            posf = 1.0f;
            const float* pd = pred_deltas + ((size_t)t * NCLS + tc) * 4;
            const float* td = tgt_deltas + (size_t)t * 4;
            #pragma unroll
            for (int k = 0; k < 4; ++k) {
                float d = fabsf(pd[k] - td[k]);
                termB += (d < 1.0f) ? 0.5f * d * d : d - 0.5f;
            }
        }
        termM = mask_partials[t];    // already masked by pos in kernel 3
    }

    __shared__ float lC[20], lB[20], lP[20], lM[20];
    float wC = wave_sum_wmma(termC);
    float wB = wave_sum_wmma(termB);
    float wP = wave_sum_wmma(posf);
    float wM = wave_sum_wmma(termM);
    int wid = t >> 5;
    if ((t & 31) == 0) { lC[wid] = wC; lB[wid] = wB; lP[wid] = wP; lM[wid] = wM; }
    __syncthreads();

    if (t == 0) {
        float sC = 0, sB = 0, sP = 0, sM = 0;
        #pragma unroll
        for (int i = 0; i < 20; ++i) { sC += lC[i]; sB += lB[i]; sP += lP[i]; sM += lM[i]; }
        out3[0] = sC / (float)R;                                 // class loss
        out3[1] = sB / fmaxf(sP * 4.0f, 1.0f);                   // bbox loss
        out3[2] = sM / fmaxf(sP * (float)MHW, 1.0f);             // mask loss
    }
}

// ---------------------------------------------------------------------------
// Launch
// ---------------------------------------------------------------------------
extern "C" void kernel_launch(void* const* d_in, const int* in_sizes, int n_in,
                              void* d_out, int out_size, void* d_ws, size_t ws_size,
                              hipStream_t stream) {
    const float* p2           = (const float*)d_in[0];
    const float* p3           = (const float*)d_in[1];
    const float* p4           = (const float*)d_in[2];
    const float* p5           = (const float*)d_in[3];
    const float* rois         = (const float*)d_in[4];
    const int*   tcid         = (const int*)  d_in[5];
    const float* pred_deltas  = (const float*)d_in[6];
    const float* tgt_deltas   = (const float*)d_in[7];
    const float* logits       = (const float*)d_in[8];
    const float* pred_masks   = (const float*)d_in[9];
    const float* target_masks = (const float*)d_in[10];

    const int N = in_sizes[4] / 5;    // 1600 ROIs
    const int R = in_sizes[5];        // 600 detections

    float* out = (float*)d_out;
    RoiParams* rp = (RoiParams*)d_ws;
    float* mpart  = (float*)((char*)d_ws + sizeof(RoiParams) * (size_t)N);

    roi_params_kernel<<<(N + 255) / 256, 256, 0, stream>>>(rois, rp, N);
    roi_align_kernel<<<N, 256, 0, stream>>>(p2, p3, p4, p5, rp, out);
    mask_partial_kernel<<<R, 256, 0, stream>>>(tcid, pred_masks, target_masks, mpart);
    losses_kernel<<<1, 640, 0, stream>>>(tcid, pred_deltas, tgt_deltas, logits,
                                         mpart, out + (size_t)N * C_CH * PP, R);
}